// cross_set_score_79096117723618
// MI455X (gfx1250) — compile-verified
//
#include <hip/hip_runtime.h>

// ---------------------------------------------------------------------------
// cross_set_score for MI455X (gfx1250, wave32, WMMA)
//
// Roofline: ~40 GFLOP of f32 matmul vs 151 MB of output. Memory floor at
// 23.3 TB/s is ~6.5us; f32 VALU would be 100s of us, so all GEMMs run on the
// matrix pipe via V_WMMA_F32_16X16X4_F32 (full f32 precision, matching the
// f32 reference). All intermediates (~25 MB) live in d_ws and fit in L2.
//
// Round-2 change: all WMMA operand fragments are fetched as aligned 8-byte
// v2f LDS loads (ds_load_b64) directly into register pairs, and the W slab
// is stored transposed in LDS, eliminating the v_dual_mov repacking chains
// seen in round 1.
// ---------------------------------------------------------------------------

#define NSET  48
#define NITEM 128
#define DIM   256
#define NHEAD 4
#define HDIM  64
#define PROJ  256   // NHEAD*HDIM

typedef float v2f __attribute__((ext_vector_type(2)));
typedef float v8f __attribute__((ext_vector_type(8)));

static __device__ __forceinline__ v8f wmma_f32_k4(v2f a, v2f b, v8f c) {
  // D = A(16x4) x B(4x16) + C, all f32.
  // 8-arg form: (neg_a, A, neg_b, B, c_mod, C, reuse_a, reuse_b)
  return __builtin_amdgcn_wmma_f32_16x16x4_f32(
      false, a, false, b, (short)0, c, false, false);
}

// ---------------------------------------------------------------------------
// Kernel 1: row-wise L2 normalization. One wave (32 lanes) per 256-elem row.
// ---------------------------------------------------------------------------
__global__ __launch_bounds__(256) void l2norm_kernel(
    const float* __restrict__ src, float* __restrict__ dst, int nrows) {
  int wave = threadIdx.x >> 5;
  int lane = threadIdx.x & 31;
  int row  = blockIdx.x * 8 + wave;
  if (row >= nrows) return;
  const float* r = src + (size_t)row * DIM;
  float v[8];
  float ss = 0.f;
#pragma unroll
  for (int i = 0; i < 8; ++i) {
    v[i] = r[lane + 32 * i];
    ss += v[i] * v[i];
  }
#pragma unroll
  for (int off = 16; off; off >>= 1) ss += __shfl_xor(ss, off, 32);
  float s = rsqrtf(fmaxf(ss, 1e-12f));
  float* o = dst + (size_t)row * DIM;
#pragma unroll
  for (int i = 0; i < 8; ++i) o[lane + 32 * i] = v[i] * s;
}

// ---------------------------------------------------------------------------
// Kernel 2: projection GEMM  P[6144x256] = S[6144x256] @ W[256x256]
// Block = 512 threads (16 waves), 128 output rows per block, K-slabs of 32.
// W slab is staged TRANSPOSED (WT[n][k], stride 36) so both A and B
// fragments are contiguous 8B v2f LDS loads. LDS = 18.4KB + 36.9KB = 55KB.
// Wave w owns row-tile (w>>1), col group (w&1)*128 .. +127 (8 tiles).
// ---------------------------------------------------------------------------
__global__ __launch_bounds__(512) void proj_kernel(
    const float* __restrict__ S, const float* __restrict__ W,
    float* __restrict__ P, int rowsTotal) {
  __shared__ __align__(16) float XS[128][36];   // S slab, row-major
  __shared__ __align__(16) float WT[256][36];   // W slab, transposed [n][k]

  int rowBase = blockIdx.x * 128;
  int t    = threadIdx.x;
  int wave = t >> 5;
  int lane = t & 31;
  int lm   = lane & 15;
  int lh   = lane >> 4;          // half-wave selector (K sub-pair)
  int m16  = (wave >> 1) * 16;   // row tile base inside block
  int nB   = (wave & 1) * 128;   // col group base

  v8f acc[8];
#pragma unroll
  for (int n = 0; n < 8; ++n) acc[n] = {};

  for (int ks = 0; ks < DIM; ks += 32) {
    // stage 128x32 slab of S (1024 float4, 2 per thread)
    for (int idx = t; idx < 1024; idx += 512) {
      int r = idx >> 3, c4 = (idx & 7) * 4;
      float4 va = *(const float4*)(S + (size_t)(rowBase + r) * DIM + ks + c4);
      *(float4*)&XS[r][c4] = va;
    }
    // stage 32x256 slab of W, transposed into WT (2048 float4 reads)
    for (int idx = t; idx < 2048; idx += 512) {
      int r = idx >> 6, c4 = (idx & 63) * 4;
      float4 vw = *(const float4*)(W + (size_t)(ks + r) * PROJ + c4);
      WT[c4 + 0][r] = vw.x;
      WT[c4 + 1][r] = vw.y;
      WT[c4 + 2][r] = vw.z;
      WT[c4 + 3][r] = vw.w;
      if (ks + 32 < DIM)
        __builtin_prefetch(W + (size_t)(ks + 32 + r) * PROJ + c4, 0, 0);
    }
    __syncthreads();
#pragma unroll
    for (int kk = 0; kk < 32; kk += 4) {
      int ka = kk + 2 * lh;
      v2f a = *(const v2f*)&XS[m16 + lm][ka];
#pragma unroll
      for (int tn = 0; tn < 8; ++tn) {
        v2f b = *(const v2f*)&WT[nB + tn * 16 + lm][ka];
        acc[tn] = wmma_f32_k4(a, b, acc[tn]);
      }
    }
    __syncthreads();
  }

  // D layout: VGPR r -> row (r + 8*lh), col = lane%16
#pragma unroll
  for (int tn = 0; tn < 8; ++tn) {
    int col = nB + tn * 16 + lm;
    int rb  = rowBase + m16 + lh * 8;
#pragma unroll
    for (int r = 0; r < 8; ++r)
      P[(size_t)(rb + r) * PROJ + col] = acc[tn][r];
  }
  (void)rowsTotal;
}

// ---------------------------------------------------------------------------
// Kernel 3: one block per (j over y-sets, i over x-sets) pair.
//  (a) cos_sim[j,i,:,:] = XN_i[128x256] @ YN_j[128x256]^T  (f32 WMMA)
//  (b) per head h: relu(LX_i_h @ LY_j_h^T) summed, combined with w2,
//      scaled by 0.125 / (nItem[i]*nItem[j]).
// 8 waves; wave w owns row tile w x all 8 col tiles (8 accumulators).
// Both operands are row-major over K, so A and B fragments are contiguous
// v2f LDS loads.
// ---------------------------------------------------------------------------
__global__ __launch_bounds__(256) void pair_kernel(
    const float* __restrict__ XN, const float* __restrict__ YN,
    const float* __restrict__ LX, const float* __restrict__ LY,
    const float* __restrict__ nItem, const float* __restrict__ w2,
    float* __restrict__ cos_out, float* __restrict__ score_out) {
  __shared__ __align__(16) float AS[128][36];
  __shared__ __align__(16) float BS[128][36];
  __shared__ float red[8];

  int pair = blockIdx.x;
  int j = pair / NSET;   // y set
  int i = pair % NSET;   // x set

  const float* Xr = XN + (size_t)i * NITEM * DIM;
  const float* Yr = YN + (size_t)j * NITEM * DIM;

  int t    = threadIdx.x;
  int wave = t >> 5;
  int lane = t & 31;
  int lm   = lane & 15;
  int lh   = lane >> 4;
  int m16  = wave * 16;

  v8f acc[8];
#pragma unroll
  for (int n = 0; n < 8; ++n) acc[n] = {};

  // ---- (a) cosine-similarity GEMM over K = 256, slabs of 32 ----
  for (int ks = 0; ks < DIM; ks += 32) {
    for (int idx = t; idx < 1024; idx += 256) {
      int r = idx >> 3, c4 = (idx & 7) * 4;
      float4 va = *(const float4*)(Xr + (size_t)r * DIM + ks + c4);
      *(float4*)&AS[r][c4] = va;
      float4 vb = *(const float4*)(Yr + (size_t)r * DIM + ks + c4);
      *(float4*)&BS[r][c4] = vb;
      if (ks + 32 < DIM) {
        __builtin_prefetch(Xr + (size_t)r * DIM + ks + 32 + c4, 0, 0);
        __builtin_prefetch(Yr + (size_t)r * DIM + ks + 32 + c4, 0, 0);
      }
    }
    __syncthreads();
#pragma unroll
    for (int kk = 0; kk < 32; kk += 4) {
      int ka = kk + 2 * lh;
      v2f a = *(const v2f*)&AS[m16 + lm][ka];
#pragma unroll
      for (int tn = 0; tn < 8; ++tn) {
        v2f b = *(const v2f*)&BS[tn * 16 + lm][ka];
        acc[tn] = wmma_f32_k4(a, b, acc[tn]);
      }
    }
    __syncthreads();
  }

  // write cos tiles: cos_out[j][i][row over x items][col over y items]
  float* C = cos_out + ((size_t)j * NSET + i) * (size_t)NITEM * NITEM;
#pragma unroll
  for (int tn = 0; tn < 8; ++tn) {
    int col = tn * 16 + lm;
    int rb  = m16 + lh * 8;
#pragma unroll
    for (int r = 0; r < 8; ++r)
      C[(size_t)(rb + r) * NITEM + col] = acc[tn][r];
  }

  // ---- (b) head attention scores: K = 64 per head, slabs of 32 ----
  const float* Ar = LX + (size_t)i * NITEM * PROJ;
  const float* Br = LY + (size_t)j * NITEM * PROJ;
  float partial = 0.f;

  for (int h = 0; h < NHEAD; ++h) {
#pragma unroll
    for (int n = 0; n < 8; ++n) acc[n] = {};
    for (int ks = 0; ks < HDIM; ks += 32) {
      int kg = h * HDIM + ks;
      for (int idx = t; idx < 1024; idx += 256) {
        int r = idx >> 3, c4 = (idx & 7) * 4;
        float4 va = *(const float4*)(Ar + (size_t)r * PROJ + kg + c4);
        *(float4*)&AS[r][c4] = va;
        float4 vb = *(const float4*)(Br + (size_t)r * PROJ + kg + c4);
        *(float4*)&BS[r][c4] = vb;
      }
      __syncthreads();
#pragma unroll
      for (int kk = 0; kk < 32; kk += 4) {
        int ka = kk + 2 * lh;
        v2f a = *(const v2f*)&AS[m16 + lm][ka];
#pragma unroll
        for (int tn = 0; tn < 8; ++tn) {
          v2f b = *(const v2f*)&BS[tn * 16 + lm][ka];
          acc[tn] = wmma_f32_k4(a, b, acc[tn]);
        }
      }
      __syncthreads();
    }
    float wh = w2[h];
#pragma unroll
    for (int tn = 0; tn < 8; ++tn)
#pragma unroll
      for (int r = 0; r < 8; ++r)
        partial += wh * fmaxf(acc[tn][r], 0.f);
  }

  // block reduction of partial: wave shuffle then LDS across 8 waves
#pragma unroll
  for (int off = 16; off; off >>= 1) partial += __shfl_xor(partial, off, 32);
  if (lane == 0) red[wave] = partial;
  __syncthreads();
  if (t == 0) {
    float s = 0.f;
#pragma unroll
    for (int w = 0; w < 8; ++w) s += red[w];
    // relu(z/8) == relu(z)/8, then / nItem[i] (x axis) / nItem[j] (y axis)
    score_out[pair] = s * 0.125f / (nItem[i] * nItem[j]);
  }
}

// ---------------------------------------------------------------------------
// Host-side launcher
// ---------------------------------------------------------------------------
extern "C" void kernel_launch(void* const* d_in, const int* in_sizes, int n_in,
                              void* d_out, int out_size, void* d_ws, size_t ws_size,
                              hipStream_t stream) {
  const float* x     = (const float*)d_in[0];  // [48,128,256]
  const float* y     = (const float*)d_in[1];  // [48,128,256]
  const float* nIt   = (const float*)d_in[2];  // [48]
  const float* W     = (const float*)d_in[3];  // [256,256]
  const float* w2    = (const float*)d_in[4];  // [4,1]

  float* out       = (float*)d_out;
  float* cos_out   = out;                                          // 48*48*128*128
  float* score_out = out + (size_t)NSET * NSET * NITEM * NITEM;    // 48*48

  const size_t MAT = (size_t)NSET * NITEM * DIM;  // 1,572,864 floats
  float* XN = (float*)d_ws;
  float* YN = XN + MAT;
  float* LX = YN + MAT;
  float* LY = LX + MAT;

  const int nrows = NSET * NITEM;  // 6144

  l2norm_kernel<<<nrows / 8, 256, 0, stream>>>(x, XN, nrows);
  l2norm_kernel<<<nrows / 8, 256, 0, stream>>>(y, YN, nrows);

  proj_kernel<<<nrows / 128, 512, 0, stream>>>(x, W, LX, nrows);
  proj_kernel<<<nrows / 128, 512, 0, stream>>>(y, W, LY, nrows);

  pair_kernel<<<NSET * NSET, 256, 0, stream>>>(XN, YN, LX, LY, nIt, w2,
                                               cos_out, score_out);
  (void)in_sizes; (void)n_in; (void)out_size; (void)ws_size;
}